// GCN_36112085025126
// MI455X (gfx1250) — compile-verified
//
#include <hip/hip_runtime.h>

typedef __attribute__((ext_vector_type(2))) float v2f;
typedef __attribute__((ext_vector_type(8))) float v8f;
typedef __attribute__((ext_vector_type(4))) unsigned v4u;
typedef __attribute__((ext_vector_type(8))) int v8i;
typedef __attribute__((ext_vector_type(4))) int v4i;
typedef __attribute__((address_space(3))) float lds_float;

// ---------------------------------------------------------------------------
// CSR build: in/out degree counts (int atomics, ~1.6M each, low contention)
// ---------------------------------------------------------------------------
__global__ __launch_bounds__(256) void count_kernel(const int* __restrict__ src,
                                                    const int* __restrict__ dst,
                                                    int* __restrict__ co,
                                                    int* __restrict__ ci, int e) {
  int t = blockIdx.x * blockDim.x + threadIdx.x;
  if (t >= e) return;
  atomicAdd(&co[src[t]], 1);
  atomicAdd(&ci[dst[t]], 1);
}

__global__ __launch_bounds__(256) void norm_from_cnt(const int* __restrict__ cnt,
                                                     float* __restrict__ nrm, int n) {
  int t = blockIdx.x * blockDim.x + threadIdx.x;
  if (t >= n) return;
  nrm[t] = rsqrtf(fmaxf((float)cnt[t], 1.0f));
}

// Single-block exclusive scan (n ~ 100K -> ~100 chunks of 1024, launched once)
__global__ __launch_bounds__(1024) void scan_kernel(const int* __restrict__ cnt,
                                                    int* __restrict__ rowptr, int n) {
  __shared__ int buf[1024];
  __shared__ int carry;
  if (threadIdx.x == 0) carry = 0;
  __syncthreads();
  for (int base = 0; base < n; base += 1024) {
    const int i = base + threadIdx.x;
    const int v = (i < n) ? cnt[i] : 0;
    buf[threadIdx.x] = v;
    __syncthreads();
    for (int off = 1; off < 1024; off <<= 1) {
      const int t = (threadIdx.x >= off) ? buf[threadIdx.x - off] : 0;
      __syncthreads();
      buf[threadIdx.x] += t;
      __syncthreads();
    }
    const int incl = buf[threadIdx.x];
    if (i < n) rowptr[i] = carry + (incl - v);
    const int tot = buf[1023];
    __syncthreads();
    if (threadIdx.x == 0) carry += tot;
    __syncthreads();
  }
  if (threadIdx.x == 0) rowptr[n] = carry;
}

__global__ __launch_bounds__(256) void fill_kernel(const int* __restrict__ src,
                                                   const int* __restrict__ dst,
                                                   int* __restrict__ cursor,
                                                   int* __restrict__ colidx, int e) {
  int t = blockIdx.x * blockDim.x + threadIdx.x;
  if (t >= e) return;
  const int p = atomicAdd(&cursor[dst[t]], 1);
  colidx[p] = src[t];
}

// ---------------------------------------------------------------------------
// GEMM: H[m,n] = sum_k A'[m,k]*W[k,n],  A'[m,k] = (X[m,k]*bna[k]+bnb[k])*nsrc[m]
// Block = 16 rows x DOUT cols (DOUT/16 waves, one 16x16 WMMA tile per wave).
// A tile (16 x DIN) staged once by the Tensor Data Mover into LDS with
// pad_enable: +4 DWORDs every 128 DWORDs -> row stride 132 floats, making the
// half-wave ds_load_b64 A-fragment reads conflict-free across all 64 banks.
// Fragment layouts per CDNA5 ISA 7.12.2 (V_WMMA_F32_16X16X4_F32):
//   A 16x4 : lane L holds row M=L&15, VGPR j holds K=(L>>4)*2+j
//   B 4x16 : lane L holds col N=L&15, VGPR j holds K=(L>>4)*2+j
//   D 16x16: VGPR v holds row M=v+(L>>4)*8, col N=L&15
// ---------------------------------------------------------------------------
template <bool BN, int DIN, int DOUT>
__global__ __launch_bounds__((DOUT / 16) * 32) void gemm_wmma(
    const float* __restrict__ X, const float* __restrict__ W,
    const float* __restrict__ nsrc, const float* __restrict__ bna,
    const float* __restrict__ bnb, float* __restrict__ H) {
  constexpr int LDSS = DIN + 4;  // padded LDS row stride (matches TDM pad)
  __shared__ float As[16 * LDSS];
  const int tid = threadIdx.x;
  const int wid = tid >> 5;
  const int l16 = tid & 15;
  const int hl = (tid >> 4) & 1;
  const int m0 = blockIdx.x << 4;

  if (wid == 0) {  // wave-uniform branch: one TDM op per block
    const unsigned lds_base = (unsigned)(unsigned long long)(lds_float*)As;
    const unsigned long long ga = (unsigned long long)(X + m0 * DIN);
    v4u g0;
    g0.x = 1u;                                    // count=1 (valid descriptor)
    g0.y = lds_base;                              // lds_addr
    g0.z = (unsigned)(ga & 0xffffffffull);        // global_addr[31:0]
    g0.w = (unsigned)((ga >> 32) & 0x1ffffffull)  // global_addr[56:32]
           | (2u << 30);                          // type=2 ("image")
    v8i g1;
    g1[0] = (int)((2u << 16)     // data_size: 4 bytes
                  | (1u << 20)   // pad_enable
                  | (6u << 22)   // pad_interval: 128 DWORDs
                  | (3u << 25)); // pad_amount: 4 DWORDs (16 B)
    g1[1] = (int)((unsigned)DIN << 16);  // tensor_dim0 = DIN (low 16)
    g1[2] = (int)(16u << 16);            // tensor_dim0 hi=0 | tensor_dim1=16 lo
    g1[3] = (int)((unsigned)DIN << 16);  // tensor_dim1 hi=0 | tile_dim0 = DIN
    g1[4] = 16;                          // tile_dim1=16, tile_dim2=0
    g1[5] = DIN;                         // tensor_dim0_stride = DIN (low 32)
    g1[6] = 0;                           // stride hi | tensor_dim1_stride lo
    g1[7] = 0;
    const v4i gz4 = {0, 0, 0, 0};             // groups 2/3 unused (2D tile)
    const v8i gz8 = {0, 0, 0, 0, 0, 0, 0, 0}; // trailing group (clang-23 form)
    __builtin_amdgcn_tensor_load_to_lds(g0, g1, gz4, gz4, gz8, 0);
    __builtin_amdgcn_s_wait_tensorcnt(0);
  }
  __syncthreads();

  const int col = (wid << 4) + l16;
  const float nm = nsrc[m0 + l16];
  v8f c = {0.f, 0.f, 0.f, 0.f, 0.f, 0.f, 0.f, 0.f};
#pragma unroll
  for (int k0 = 0; k0 < DIN; k0 += 4) {
    const int k = k0 + (hl << 1);
    const float2 ap = *(const float2*)&As[l16 * LDSS + k];  // ds_load_b64
    float a0 = ap.x, a1 = ap.y;
    if constexpr (BN) {  // fused BatchNorm affine (branchless)
      a0 = fmaf(a0, bna[k], bnb[k]);
      a1 = fmaf(a1, bna[k + 1], bnb[k + 1]);
    }
    const v2f av = {a0 * nm, a1 * nm};
    const v2f bv = {W[k * DOUT + col], W[(k + 1) * DOUT + col]};
    c = __builtin_amdgcn_wmma_f32_16x16x4_f32(false, av, false, bv, (short)0, c,
                                              false, false);
  }
#pragma unroll
  for (int v = 0; v < 8; ++v)
    H[(m0 + v + (hl << 3)) * DOUT + col] = c[v];
}

// ---------------------------------------------------------------------------
// CSR gather-aggregate with fused epilogue: one wave per node, each lane owns
// D/32 features in registers; b128 gathers of L2-resident H rows, ZERO atomics.
// out[v,:] = maybe_relu( (sum_{e: dst=v} H[src_e,:]) * ndst[v] + bias )
// ---------------------------------------------------------------------------
template <int D, bool RELU>
__global__ __launch_bounds__(256) void gather_kernel(
    const float* __restrict__ H, const int* __restrict__ rowptr,
    const int* __restrict__ colidx, const float* __restrict__ ndst,
    const float* __restrict__ bias, float* __restrict__ out, int n) {
  constexpr int VEC = D / 32;  // 4 (D=128) or 2 (D=64)
  const int wave = (blockIdx.x * blockDim.x + threadIdx.x) >> 5;
  const int lane = threadIdx.x & 31;
  if (wave >= n) return;
  const int beg = rowptr[wave];
  const int end = rowptr[wave + 1];
  const int f = lane * VEC;
  float4 acc = {0.f, 0.f, 0.f, 0.f};
  for (int i = beg; i < end; ++i) {
    __builtin_prefetch(&colidx[i + 16], 0, 1);  // global_prefetch of index stream
    const int s = colidx[i];
    const float* hp = H + s * D + f;
    if constexpr (VEC == 4) {
      const float4 hv = *(const float4*)hp;
      acc.x += hv.x; acc.y += hv.y; acc.z += hv.z; acc.w += hv.w;
    } else {
      const float2 hv = *(const float2*)hp;
      acc.x += hv.x; acc.y += hv.y;
    }
  }
  const float nm = ndst[wave];
  float* op = out + wave * D + f;
  if constexpr (VEC == 4) {
    float4 r;
    r.x = fmaf(acc.x, nm, bias[f + 0]);
    r.y = fmaf(acc.y, nm, bias[f + 1]);
    r.z = fmaf(acc.z, nm, bias[f + 2]);
    r.w = fmaf(acc.w, nm, bias[f + 3]);
    if constexpr (RELU) {
      r.x = fmaxf(r.x, 0.f); r.y = fmaxf(r.y, 0.f);
      r.z = fmaxf(r.z, 0.f); r.w = fmaxf(r.w, 0.f);
    }
    *(float4*)op = r;
  } else {
    float2 r;
    r.x = fmaf(acc.x, nm, bias[f + 0]);
    r.y = fmaf(acc.y, nm, bias[f + 1]);
    if constexpr (RELU) { r.x = fmaxf(r.x, 0.f); r.y = fmaxf(r.y, 0.f); }
    *(float2*)op = r;
  }
}

// Per-feature sum / sum-of-squares; register accumulation + 2 atomics/block.
__global__ void bn_stats_kernel(const float* __restrict__ y,
                                float* __restrict__ sum,
                                float* __restrict__ sumsq, int n, int d) {
  const int f = threadIdx.x;
  const int chunk = (n + gridDim.x - 1) / gridDim.x;
  const int start = blockIdx.x * chunk;
  const int end = min(start + chunk, n);
  float s = 0.f, ss = 0.f;
  for (int i = start; i < end; ++i) {
    const float v = y[i * d + f];
    s += v;
    ss = fmaf(v, v, ss);
  }
  unsafeAtomicAdd(&sum[f], s);
  unsafeAtomicAdd(&sumsq[f], ss);
}

// Collapse BN to per-feature affine, fused into the next GEMM's A-load.
__global__ void bn_finalize_kernel(const float* __restrict__ sum,
                                   const float* __restrict__ sumsq,
                                   const float* __restrict__ gamma,
                                   const float* __restrict__ beta,
                                   float* __restrict__ bna,
                                   float* __restrict__ bnb, float invn) {
  const int f = threadIdx.x;
  const float mu = sum[f] * invn;
  const float var = fmaf(-mu, mu, sumsq[f] * invn);
  const float a = rsqrtf(var + 1e-5f) * gamma[f];
  bna[f] = a;
  bnb[f] = fmaf(-mu, a, beta[f]);
}

// ---------------------------------------------------------------------------
extern "C" void kernel_launch(void* const* d_in, const int* in_sizes, int n_in,
                              void* d_out, int out_size, void* d_ws, size_t ws_size,
                              hipStream_t stream) {
  const float* feats = (const float*)d_in[0];
  const int* src = (const int*)d_in[1];
  const int* dst = (const int*)d_in[2];
  const float* W1 = (const float*)d_in[3];
  const float* b1 = (const float*)d_in[4];
  const float* gam1 = (const float*)d_in[5];
  const float* bet1 = (const float*)d_in[6];
  const float* W2 = (const float*)d_in[7];
  const float* b2 = (const float*)d_in[8];
  const float* W3 = (const float*)d_in[9];
  const float* b3 = (const float*)d_in[10];
  float* out = (float*)d_out;

  constexpr int DH = 128, DO = 64, DI = 128;
  const int N = in_sizes[0] / DI;   // 100000
  const int E = in_sizes[1];        // 1600000

  // ---- workspace layout -------------------------------------------------
  char* ws = (char*)d_ws;
  size_t off = 0;
  auto carve = [&](size_t bytes) -> void* {
    void* p = ws + off;
    off += (bytes + 255) & ~(size_t)255;
    return p;
  };
  float* nsrc = (float*)carve((size_t)N * 4);
  float* ndst = (float*)carve((size_t)N * 4);
  int* cnt_o  = (int*)carve((size_t)N * 4);
  int* cnt_i  = (int*)carve((size_t)N * 4);
  int* rowptr = (int*)carve((size_t)(N + 1) * 4);
  int* cursor = (int*)carve((size_t)N * 4);
  int* colidx = (int*)carve((size_t)E * 4);
  float* h    = (float*)carve((size_t)N * DH * 4);
  float* xb   = (float*)carve((size_t)N * DH * 4);
  float* bnsum = (float*)carve((size_t)DH * 4);
  float* bnss  = (float*)carve((size_t)DH * 4);
  float* bna   = (float*)carve((size_t)DH * 4);
  float* bnb   = (float*)carve((size_t)DH * 4);
  (void)ws_size; (void)n_in; (void)out_size;

  const int nodeBlk = (N + 255) / 256;
  const int edgeBlk = (E + 255) / 256;
  const int gathBlk = (N * 32 + 255) / 256;  // one wave per node

  // ---- CSR build + norms (shared by all three convs) --------------------
  (void)hipMemsetAsync(cnt_o, 0, (size_t)N * 4, stream);
  (void)hipMemsetAsync(cnt_i, 0, (size_t)N * 4, stream);
  count_kernel<<<edgeBlk, 256, 0, stream>>>(src, dst, cnt_o, cnt_i, E);
  norm_from_cnt<<<nodeBlk, 256, 0, stream>>>(cnt_o, nsrc, N);
  norm_from_cnt<<<nodeBlk, 256, 0, stream>>>(cnt_i, ndst, N);
  scan_kernel<<<1, 1024, 0, stream>>>(cnt_i, rowptr, N);
  (void)hipMemcpyAsync(cursor, rowptr, (size_t)N * 4, hipMemcpyDeviceToDevice,
                       stream);
  fill_kernel<<<edgeBlk, 256, 0, stream>>>(src, dst, cursor, colidx, E);

  // ---- layer 1: conv -> relu (xb), then BN stats ------------------------
  gemm_wmma<false, DI, DH><<<N / 16, (DH / 16) * 32, 0, stream>>>(
      feats, W1, nsrc, nullptr, nullptr, h);
  gather_kernel<DH, true><<<gathBlk, 256, 0, stream>>>(h, rowptr, colidx, ndst,
                                                       b1, xb, N);
  (void)hipMemsetAsync(bnsum, 0, (size_t)DH * 4, stream);
  (void)hipMemsetAsync(bnss, 0, (size_t)DH * 4, stream);
  bn_stats_kernel<<<200, DH, 0, stream>>>(xb, bnsum, bnss, N, DH);
  bn_finalize_kernel<<<1, DH, 0, stream>>>(bnsum, bnss, gam1, bet1, bna, bnb,
                                           1.0f / (float)N);

  // ---- layer 2: BN fused into A-load, conv -> relu ----------------------
  gemm_wmma<true, DH, DH><<<N / 16, (DH / 16) * 32, 0, stream>>>(
      xb, W2, nsrc, bna, bnb, h);
  gather_kernel<DH, true><<<gathBlk, 256, 0, stream>>>(h, rowptr, colidx, ndst,
                                                       b2, xb, N);

  // ---- layer 3: conv (no relu) -> final output --------------------------
  gemm_wmma<false, DH, DO><<<N / 16, (DO / 16) * 32, 0, stream>>>(
      xb, W3, nsrc, nullptr, nullptr, h);
  gather_kernel<DO, false><<<gathBlk, 256, 0, stream>>>(h, rowptr, colidx, ndst,
                                                        b3, out, N);
}